// GNN_77068893160011
// MI455X (gfx1250) — compile-verified
//
#include <hip/hip_runtime.h>

typedef __attribute__((ext_vector_type(2))) float v2f;
typedef __attribute__((ext_vector_type(8))) float v8f;

// ---------------- degree / norm kernels ----------------

__global__ __launch_bounds__(256) void k_fill_ones(float* __restrict__ deg, int n) {
    int i = blockIdx.x * 256 + threadIdx.x;
    if (i < n) deg[i] = 1.0f;   // self-loop contributes 1 to every node's degree
}

__global__ __launch_bounds__(256) void k_deg_count(const int* __restrict__ dst,
                                                   float* __restrict__ deg, int E) {
    int e = blockIdx.x * 256 + threadIdx.x;
    if (e < E) atomicAdd(&deg[dst[e]], 1.0f);
}

__global__ __launch_bounds__(256) void k_rsqrt_inplace(float* __restrict__ deg, int n) {
    int i = blockIdx.x * 256 + threadIdx.x;
    if (i < n) deg[i] = rsqrtf(deg[i]);  // deg >= 1 always, matches where(deg>0,...)
}

// ---------------- fp32 WMMA GEMM: H[M x NC] = X[M x K] * W[K x NC] ----------------
// One wave computes one 16x16 output tile via a chain of V_WMMA_F32_16X16X4_F32.
// A (16x4, 32-bit): lanes 0-15 hold M=0..15 with K=0,1 in the two VGPRs;
//                   lanes 16-31 hold K=2,3.   B (4x16) mirrors this for N.
// C/D (16x16, 32-bit): VGPR j = row j (lanes 0-15) / row j+8 (lanes 16-31).

template <int K, int NC>
__global__ __launch_bounds__(256) void k_gemm_wmma(const float* __restrict__ X,
                                                   const float* __restrict__ W,
                                                   float* __restrict__ H, int M) {
    constexpr int tilesN = NC / 16;
    const int lane = threadIdx.x & 31;
    const int wave = threadIdx.x >> 5;
    const int tile = blockIdx.x * 8 + wave;
    const int totalTiles = (M >> 4) * tilesN;
    if (tile >= totalTiles) return;              // wave-uniform: EXEC stays all-ones
    const int tm = tile / tilesN;
    const int tn = tile - tm * tilesN;

    const int mrow = tm * 16 + (lane & 15);
    const int koff = (lane >> 4) * 2;            // 0 for lanes 0-15, 2 for lanes 16-31
    const int ncol = tn * 16 + (lane & 15);

    v8f acc = {};
#pragma unroll 4
    for (int k0 = 0; k0 < K; k0 += 4) {
        v2f a, b;
        a.x = X[mrow * K + k0 + koff];
        a.y = X[mrow * K + k0 + koff + 1];
        b.x = W[(k0 + koff) * NC + ncol];
        b.y = W[(k0 + koff + 1) * NC + ncol];
        acc = __builtin_amdgcn_wmma_f32_16x16x4_f32(
            /*neg_a=*/false, a, /*neg_b=*/false, b,
            /*c_mod=*/(short)0, acc, /*reuse_a=*/false, /*reuse_b=*/false);
    }

    const int rbase = tm * 16 + (lane >> 4) * 8;
    const int c     = tn * 16 + (lane & 15);
#pragma unroll
    for (int j = 0; j < 8; ++j)
        H[(rbase + j) * NC + c] = acc[j];
}

// ---------------- self-loop + bias init: out = bias + h * dinv^2 ----------------

template <int DIM>
__global__ __launch_bounds__(256) void k_selfloop_bias(const float* __restrict__ h,
                                                       const float* __restrict__ bias,
                                                       const float* __restrict__ dinv,
                                                       float* __restrict__ out, int n) {
    int idx = blockIdx.x * 256 + threadIdx.x;
    if (idx >= n * DIM) return;
    int i = idx / DIM;
    int d = idx - i * DIM;
    float di = dinv[i];
    out[idx] = bias[d] + h[idx] * di * di;
}

// ---------------- edge scatter: out[dst] += h[src] * dinv[src]*dinv[dst] ----------------
// float4 gather per thread, 4 float atomics; DIM/4 threads cooperate per edge.

template <int DIM>
__global__ __launch_bounds__(256) void k_edge_scatter(const float* __restrict__ h,
                                                      const int* __restrict__ src,
                                                      const int* __restrict__ dst,
                                                      const float* __restrict__ dinv,
                                                      float* __restrict__ out, int E) {
    constexpr int PER = DIM / 4;
    long long idx = (long long)blockIdx.x * 256 + threadIdx.x;
    int e = (int)(idx / PER);
    if (e >= E) return;
    int q = (int)(idx - (long long)e * PER) * 4;
    int s = src[e], t = dst[e];
    float nrm = dinv[s] * dinv[t];
    const float4 v = *(const float4*)(h + (size_t)s * DIM + q);
    float* o = out + (size_t)t * DIM + q;
    atomicAdd(o + 0, v.x * nrm);
    atomicAdd(o + 1, v.y * nrm);
    atomicAdd(o + 2, v.z * nrm);
    atomicAdd(o + 3, v.w * nrm);
}

__global__ __launch_bounds__(256) void k_relu(float* __restrict__ x, int n) {
    int i = blockIdx.x * 256 + threadIdx.x;
    if (i < n) x[i] = fmaxf(x[i], 0.0f);
}

// ---------------- launch ----------------

extern "C" void kernel_launch(void* const* d_in, const int* in_sizes, int n_in,
                              void* d_out, int out_size, void* d_ws, size_t ws_size,
                              hipStream_t stream) {
    const float* x  = (const float*)d_in[0];
    const int*   ei = (const int*)d_in[1];
    const float* W1 = (const float*)d_in[2];
    const float* b1 = (const float*)d_in[3];
    const float* W2 = (const float*)d_in[4];
    const float* b2 = (const float*)d_in[5];
    const float* W3 = (const float*)d_in[6];
    const float* b3 = (const float*)d_in[7];

    constexpr int IN = 128, HID = 64, OUT = 112;
    const int N = in_sizes[0] / IN;     // 100000
    const int E = in_sizes[1] / 2;      // 1600000
    const int* src = ei;
    const int* dst = ei + E;

    // workspace layout: [dinv: N f32][A: N*OUT f32 (h buffer)][B: N*HID f32 (agg buffer)]
    char* ws = (char*)d_ws;
    float* dinv = (float*)ws;
    size_t off = ((size_t)N * 4 + 255) & ~(size_t)255;
    float* A = (float*)(ws + off);
    off += ((size_t)N * OUT * 4 + 255) & ~(size_t)255;
    float* B = (float*)(ws + off);
    float* out = (float*)d_out;

    auto cdiv = [](long long a, long long b) { return (int)((a + b - 1) / b); };

    // --- degree / normalization (dinv recomputed every call: deterministic) ---
    k_fill_ones<<<cdiv(N, 256), 256, 0, stream>>>(dinv, N);
    k_deg_count<<<cdiv(E, 256), 256, 0, stream>>>(dst, dinv, E);
    k_rsqrt_inplace<<<cdiv(N, 256), 256, 0, stream>>>(dinv, N);

    // --- layer 1: h1 = x @ W1 -> A ; agg -> B ; relu ---
    k_gemm_wmma<IN, HID><<<cdiv((long long)(N / 16) * (HID / 16), 8), 256, 0, stream>>>(x, W1, A, N);
    k_selfloop_bias<HID><<<cdiv((long long)N * HID, 256), 256, 0, stream>>>(A, b1, dinv, B, N);
    k_edge_scatter<HID><<<cdiv((long long)E * (HID / 4), 256), 256, 0, stream>>>(A, src, dst, dinv, B, E);
    k_relu<<<cdiv((long long)N * HID, 256), 256, 0, stream>>>(B, N * HID);

    // --- layer 2: h2 = B @ W2 -> A ; agg -> B ; relu ---
    k_gemm_wmma<HID, HID><<<cdiv((long long)(N / 16) * (HID / 16), 8), 256, 0, stream>>>(B, W2, A, N);
    k_selfloop_bias<HID><<<cdiv((long long)N * HID, 256), 256, 0, stream>>>(A, b2, dinv, B, N);
    k_edge_scatter<HID><<<cdiv((long long)E * (HID / 4), 256), 256, 0, stream>>>(A, src, dst, dinv, B, E);
    k_relu<<<cdiv((long long)N * HID, 256), 256, 0, stream>>>(B, N * HID);

    // --- layer 3: h3 = B @ W3 -> A ; agg -> d_out (no relu) ---
    k_gemm_wmma<HID, OUT><<<cdiv((long long)(N / 16) * (OUT / 16), 8), 256, 0, stream>>>(B, W3, A, N);
    k_selfloop_bias<OUT><<<cdiv((long long)N * OUT, 256), 256, 0, stream>>>(A, b3, dinv, out, N);
    k_edge_scatter<OUT><<<cdiv((long long)E * (OUT / 4), 256), 256, 0, stream>>>(A, src, dst, dinv, out, E);
}